// MothFactorMatrix_75333726372115
// MI455X (gfx1250) — compile-verified
//
#include <hip/hip_runtime.h>

typedef __attribute__((ext_vector_type(2))) float v2f;
typedef __attribute__((ext_vector_type(8))) float v8f;

#define NCOL   4096            // N
#define KDIM   128             // per-factor contiguous block
#define BDIM   32              // B (32x32 per-diagonal matrices)
#define DDIM   4               // D (diagonal length)
#define WAVES  4               // waves per block (wave32)
#define TROWS  16              // lead rows per wave tile (WMMA M)
#define XMS    36              // x/out d-plane row stride in floats (32 + 4 pad: conflict-free b64 gathers)
#define XPL    (TROWS * XMS)   // 576 floats: one d-plane of a wave's x tile
#define XSL    (DDIM * XPL)    // 2304 floats: per-wave slice (reused for out tile)
#define MMS    36              // mat d-plane row stride
#define MPL    (BDIM * MMS)    // 1152 floats: one d-plane of mat[f]

__global__ __launch_bounds__(WAVES * 32) void moth_wmma_kernel(
    const float* __restrict__ x, const float* __restrict__ mat,
    float* __restrict__ out, int nrows)
{
    __shared__ float ml2[DDIM * MPL];        // mat[f] as 4 d-planes of 32x32 (+pad)   18.4 KB
    __shared__ float xsb[WAVES * XSL];       // per-wave x tile as 4 d-planes of 16x32 36.9 KB

    const int lane = threadIdx.x & 31;
    const int wave = threadIdx.x >> 5;
    const int f    = blockIdx.y;
    const int tile = blockIdx.x * WAVES + wave;

    // ---- stage mat[f]: 1024 float4 loads split across 128 threads; scatter into d-planes ----
    const float4* mf4 = (const float4*)(mat + (size_t)f * (BDIM * KDIM));
    #pragma unroll
    for (int j = 0; j < 8; ++j) {
        int i4 = j * 128 + threadIdx.x;      // coalesced float4 index, 0..1023
        float4 v = mf4[i4];
        int r = i4 >> 5, c = i4 & 31;        // mat[f, r, c, d] = v[d]
        ml2[0 * MPL + r * MMS + c] = v.x;
        ml2[1 * MPL + r * MMS + c] = v.y;
        ml2[2 * MPL + r * MMS + c] = v.z;
        ml2[3 * MPL + r * MMS + c] = v.w;
    }
    __syncthreads();

    const size_t L0 = (size_t)tile * TROWS;
    if (L0 >= (size_t)nrows) return;         // wave-uniform; no barriers after this point

    // ---- stage this wave's 16x128 x tile into d-planes (coalesced float4 global loads) ----
    float* xs = &xsb[wave * XSL];
    const float* xw = x + L0 * NCOL + (size_t)f * KDIM;
    #pragma unroll
    for (int row = 0; row < TROWS; ++row) {
        const float4 v = *(const float4*)(xw + (size_t)row * NCOL + lane * 4);
        // lane's float4 covers (c = lane, d = 0..3); conflict-free scatter stores
        xs[0 * XPL + row * XMS + lane] = v.x;
        xs[1 * XPL + row * XMS + lane] = v.y;
        xs[2 * XPL + row * XMS + lane] = v.z;
        xs[3 * XPL + row * XMS + lane] = v.w;
    }

    // ---- WMMA: per (d, r-half): D(16L x 16r) += A(16L x 4c) x B(4c x 16r), 8 c-chunks ----
    const int arow = lane & 15;              // A: M row / B: N col (lanes mod 16)
    const int kh   = (lane >> 4) << 1;       // K sub-offset: 0 (lanes 0-15), 2 (lanes 16-31)

    v8f acc[DDIM][2];
    #pragma unroll
    for (int d = 0; d < DDIM; ++d)
        #pragma unroll
        for (int rt = 0; rt < 2; ++rt) {
            v8f z = {0.f, 0.f, 0.f, 0.f, 0.f, 0.f, 0.f, 0.f};
            acc[d][rt] = z;
        }

    #pragma unroll
    for (int k = 0; k < 8; ++k) {
        #pragma unroll
        for (int d = 0; d < DDIM; ++d) {
            // adjacent pair -> single ds_load_b64 into an even-aligned VGPR pair
            const float2 a = *(const float2*)&xs[d * XPL + arow * XMS + 4 * k + kh];
            v2f A = {a.x, a.y};
            #pragma unroll
            for (int rt = 0; rt < 2; ++rt) {
                const float2 b = *(const float2*)&ml2[d * MPL + (rt * 16 + arow) * MMS + 4 * k + kh];
                v2f B = {b.x, b.y};
                acc[d][rt] = __builtin_amdgcn_wmma_f32_16x16x4_f32(
                    false, A, false, B, (short)0, acc[d][rt], false, false);
            }
        }
    }

    // ---- scatter D tiles into this wave's slice as out d-planes [d][m][r] (conflict-free) ----
    const int mofs = (lane >> 4) << 3;       // lanes 16-31 hold M = v + 8
    #pragma unroll
    for (int d = 0; d < DDIM; ++d)
        #pragma unroll
        for (int rt = 0; rt < 2; ++rt)
            #pragma unroll
            for (int v = 0; v < 8; ++v)
                xs[d * XPL + (v + mofs) * XMS + rt * 16 + arow] = acc[d][rt][v];

    // ---- gather 4 d-planes -> float4, coalesced global writeback ----
    float* ow = out + L0 * NCOL + (size_t)f * KDIM;
    #pragma unroll
    for (int row = 0; row < TROWS; ++row) {
        float4 q;
        q.x = xs[0 * XPL + row * XMS + lane];
        q.y = xs[1 * XPL + row * XMS + lane];
        q.z = xs[2 * XPL + row * XMS + lane];
        q.w = xs[3 * XPL + row * XMS + lane];
        *(float4*)(ow + (size_t)row * NCOL + lane * 4) = q;
    }
}

extern "C" void kernel_launch(void* const* d_in, const int* in_sizes, int n_in,
                              void* d_out, int out_size, void* d_ws, size_t ws_size,
                              hipStream_t stream) {
    const float* x   = (const float*)d_in[0];   // (4, 4096, 4096) f32
    const float* mat = (const float*)d_in[1];   // (32, 32, 32, 4) f32
    float* out = (float*)d_out;

    const int nrows = in_sizes[0] / NCOL;       // 16384 lead rows
    const int tiles = nrows / TROWS;            // 1024 row tiles
    dim3 grid(tiles / WAVES, 32, 1);            // (256, F=32)
    dim3 block(WAVES * 32, 1, 1);               // 128 threads = 4 wave32
    hipLaunchKernelGGL(moth_wmma_kernel, grid, block, 0, stream, x, mat, out, nrows);
}